// EdgeCormorant_32478542692892
// MI455X (gfx1250) — compile-verified
//
#include <hip/hip_runtime.h>
#include <math.h>

// ---------------- problem constants (match reference) ----------------
constexpr int BB    = 4;
constexpr int NN    = 96;
constexpr int CC    = 64;
constexpr int NBAS  = 32;
constexpr int BN    = BB * NN;        // 384
constexpr int EE    = BB * NN * NN;   // 36864 edges
constexpr float RMAXF   = 3.0f;
constexpr float SOFTW   = 0.2f;
constexpr float CHSCALE = 9.0f;

typedef float v2f __attribute__((ext_vector_type(2)));
typedef float v8f __attribute__((ext_vector_type(8)));

// ============================================================
// Clebsch-Gordan table (Racah formula), computed on device.
// cgtab index: ((((l1*3+l2)*3+L)*5+m1i)*5+m2i)*5+moi
// ============================================================
__device__ __forceinline__ double dfact(int n) {
    double r = 1.0;
    for (int i = 2; i <= n; ++i) r *= (double)i;
    return r;
}

__global__ void cg_init_kernel(float* cgtab) {
    int gid = blockIdx.x * blockDim.x + threadIdx.x;
    const int TOT = 3 * 3 * 3 * 5 * 5 * 5;
    if (gid >= TOT) return;
    int t   = gid;
    int moi = t % 5; t /= 5;
    int m2i = t % 5; t /= 5;
    int m1i = t % 5; t /= 5;
    int L   = t % 3; t /= 3;
    int l2  = t % 3; t /= 3;
    int l1  = t;
    float val = 0.0f;
    int dl = (l1 > l2) ? (l1 - l2) : (l2 - l1);
    if (m1i <= 2 * l1 && m2i <= 2 * l2 && moi <= 2 * L && L >= dl && L <= l1 + l2) {
        int m1 = m1i - l1, m2 = m2i - l2, M = moi - L;
        if (M == m1 + m2) {
            double pre = sqrt((2.0 * L + 1.0) * dfact(L + l1 - l2) * dfact(L - l1 + l2) *
                              dfact(l1 + l2 - L) / dfact(l1 + l2 + L + 1));
            pre *= sqrt(dfact(L + M) * dfact(L - M) * dfact(l1 - m1) * dfact(l1 + m1) *
                        dfact(l2 - m2) * dfact(l2 + m2));
            double s = 0.0;
            for (int k = 0; k <= l1 + l2 - L; ++k) {
                int d0 = k, d1 = l1 + l2 - L - k, d2 = l1 - m1 - k;
                int d3 = l2 + m2 - k, d4 = L - l2 + m1 + k, d5 = L - l1 - m2 + k;
                if (d0 < 0 || d1 < 0 || d2 < 0 || d3 < 0 || d4 < 0 || d5 < 0) continue;
                double den = dfact(d0) * dfact(d1) * dfact(d2) * dfact(d3) * dfact(d4) * dfact(d5);
                s += ((k & 1) ? -1.0 : 1.0) / den;
            }
            val = (float)(pre * s);
        }
    }
    cgtab[gid] = val;
}

// ============================================================
// Geometry: spherical harmonics (9 complex slots), cutoff, radial
// basis [E,32]; also zero-init pred accumulator.
// sph layout: [e][slot 0..8][re,im]
// ============================================================
__global__ void geom_kernel(const float* __restrict__ pos, const float* __restrict__ am,
                            float* __restrict__ sph, float* __restrict__ cut,
                            float* __restrict__ basis, float* __restrict__ pred) {
    int e = blockIdx.x * blockDim.x + threadIdx.x;
    if (e >= EE) return;
    int j = e % NN, bi = e / NN;
    int i = bi % NN, b = bi / NN;
    float px = pos[(b * NN + i) * 3 + 0] - pos[(b * NN + j) * 3 + 0];
    float py = pos[(b * NN + i) * 3 + 1] - pos[(b * NN + j) * 3 + 1];
    float pz = pos[(b * NN + i) * 3 + 2] - pos[(b * NN + j) * 3 + 2];
    float d2 = px * px + py * py + pz * pz;
    bool ok = (d2 > 1e-12f);
    float mask  = am[b * NN + i] * am[b * NN + j];
    float valid = ok ? mask : 0.0f;
    float nrm = ok ? sqrtf(d2) : 0.0f;
    float inv = ok ? (1.0f / sqrtf(d2)) : 1.0f;
    float x = px * inv, y = py * inv, z = pz * inv;
    const float c1  = 0.70710678118654752f;   // 1/sqrt(2)
    const float c2a = 0.61237243569579452f;   // sqrt(3/8)
    const float c2b = 1.22474487139158905f;   // sqrt(1.5)
    float* S = sph + (long)e * 18;
    // l = 0
    S[0] = valid;            S[1] = 0.0f;
    // l = 1
    S[2] = c1 * x * valid;   S[3] = -c1 * y * valid;
    S[4] = z * valid;        S[5] = 0.0f;
    S[6] = -c1 * x * valid;  S[7] = -c1 * y * valid;
    // l = 2
    float xx = x * x, yy = y * y, zz = z * z;
    S[8]  =  c2a * (xx - yy) * valid;      S[9]  = -2.0f * c2a * x * y * valid;
    S[10] =  c2b * z * x * valid;          S[11] = -c2b * z * y * valid;
    S[12] =  0.5f * (3.0f * zz - 1.0f) * valid; S[13] = 0.0f;
    S[14] = -c2b * z * x * valid;          S[15] = -c2b * z * y * valid;
    S[16] =  c2a * (xx - yy) * valid;      S[17] =  2.0f * c2a * x * y * valid;
    // cutoff
    float sig = 1.0f / (1.0f + expf(-(RMAXF - nrm) / SOFTW));
    cut[e] = valid * ((nrm < RMAXF) ? 1.0f : 0.0f) * sig;
    // radial basis
    const float width = RMAXF / (float)NBAS;
    for (int k = 0; k < NBAS; ++k) {
        float ck = RMAXF * (float)k / (float)(NBAS - 1);
        float dn = (nrm - ck) / width;
        basis[(long)e * NBAS + k] = expf(-dn * dn);
    }
    pred[e] = 0.0f;
}

// ============================================================
// Input featurization: atom[l=0] = (scalars @ W_in) * mask (real part).
// atom layout: [bn][slot 0..8][c][re,im], slot(l,m) = l*l + m
// ============================================================
__global__ void input_kernel(const int* __restrict__ species, const int* __restrict__ charges,
                             const float* __restrict__ am, const float* __restrict__ Win,
                             float* __restrict__ atom) {
    int gid = blockIdx.x * blockDim.x + threadIdx.x;
    if (gid >= BN * CC) return;
    int c = gid % CC, bn = gid / CC;
    float q = (float)charges[bn] / CHSCALE;
    int sp = species[bn];
    float acc = 0.0f, qp = 1.0f;
    for (int p = 0; p <= 2; ++p) { acc += qp * Win[(sp * 3 + p) * CC + c]; qp *= q; }
    acc *= am[bn];
    for (int s = 0; s < 9; ++s) {
        long o = (((long)bn * 9 + s) * CC + c) * 2;
        atom[o] = 0.0f; atom[o + 1] = 0.0f;
    }
    atom[(((long)bn * 9) * CC + c) * 2] = acc;
}

// ============================================================
// Pairwise dots: dots[e][t][l'*C + c] = sum_m conj-dot of atom reps.
// dots row stride = 3C (re/im rows interleaved: row r = e*2+t).
// ============================================================
__global__ void dots_kernel(const float* __restrict__ atom, float* __restrict__ dots, int n_l) {
    int gid = blockIdx.x * blockDim.x + threadIdx.x;
    if (gid >= EE * CC) return;
    int c = gid & (CC - 1);
    int e = gid >> 6;
    int j = e % NN, bi = e / NN;
    int i = bi % NN, b = bi / NN;
    long bni = (long)(b * NN + i), bnj = (long)(b * NN + j);
    for (int l = 0; l < n_l; ++l) {
        float re = 0.0f, im = 0.0f;
        int base = l * l;
        for (int m = 0; m <= 2 * l; ++m) {
            const float* Ai = atom + ((bni * 9 + base + m) * CC + c) * 2;
            const float* Aj = atom + ((bnj * 9 + base + m) * CC + c) * 2;
            float ar = Ai[0], ai = Ai[1], br = Aj[0], bi2 = Aj[1];
            re += ar * br + ai * bi2;
            im += ai * br - ar * bi2;
        }
        dots[((long)e * 2 + 0) * (3 * CC) + l * CC + c] = re;
        dots[((long)e * 2 + 1) * (3 * CC) + l * CC + c] = im;
    }
}

// ============================================================
// WMMA fp32 GEMM: radial weights, all 3 l at once (grid.y = l).
// rad3[l][e][c] = basis[e][:] @ WradLev[l][:,c], K = 32.
// ============================================================
__global__ void __launch_bounds__(32)
rad3_wmma_kernel(const float* __restrict__ basis, const float* __restrict__ WradLev,
                 float* __restrict__ rad3) {
    int lane = threadIdx.x;
    int hi = lane >> 4, lr = lane & 15;
    int l = blockIdx.y;
    const float* W = WradLev + (long)l * NBAS * CC;
    float* rad = rad3 + (long)l * EE * CC;
    long r = (long)blockIdx.x * 16 + lr;
    v8f acc0 = {0,0,0,0,0,0,0,0}, acc1 = acc0, acc2 = acc0, acc3 = acc0;
    const float* Ar = basis + r * NBAS;
    for (int k0 = 0; k0 < NBAS; k0 += 4) {
        v2f a; a.x = Ar[k0 + 2 * hi]; a.y = Ar[k0 + 2 * hi + 1];
        const float* Wp = W + (long)(k0 + 2 * hi) * CC + lr;
        v2f b0; b0.x = Wp[0];  b0.y = Wp[CC];
        v2f b1; b1.x = Wp[16]; b1.y = Wp[CC + 16];
        v2f b2; b2.x = Wp[32]; b2.y = Wp[CC + 32];
        v2f b3; b3.x = Wp[48]; b3.y = Wp[CC + 48];
        acc0 = __builtin_amdgcn_wmma_f32_16x16x4_f32(false, a, false, b0, (short)0, acc0, false, false);
        acc1 = __builtin_amdgcn_wmma_f32_16x16x4_f32(false, a, false, b1, (short)0, acc1, false, false);
        acc2 = __builtin_amdgcn_wmma_f32_16x16x4_f32(false, a, false, b2, (short)0, acc2, false, false);
        acc3 = __builtin_amdgcn_wmma_f32_16x16x4_f32(false, a, false, b3, (short)0, acc3, false, false);
    }
    long rbase = (long)blockIdx.x * 16;
#pragma unroll
    for (int v = 0; v < 8; ++v) {
        long row = rbase + v + 8 * hi;
        rad[row * CC + 0  + lr] = acc0[v];
        rad[row * CC + 16 + lr] = acc1[v];
        rad[row * CC + 32 + lr] = acc2[v];
        rad[row * CC + 48 + lr] = acc3[v];
    }
}

// ============================================================
// WMMA fp32 GEMM: fused edge mixing for all 3 l.
// Rows r = e*2 + t (73728). feats row(l) = [prevE_l (w0)] ++ [dots (w1)].
// The dots segment of A is shared across l -> loaded once, feeds
// 12 accumulators (3 l x 4 col tiles = full 16x64 per l).
// edgeNew[l][e][t][c] = (acc * inv_sqrt_k) * rad3[l][e][c] * cut[e]
// ============================================================
__global__ void __launch_bounds__(32)
mix3_wmma_kernel(const float* __restrict__ prevE, int w0,
                 const float* __restrict__ dots, int w1,
                 const float* __restrict__ W0, const float* __restrict__ W1,
                 const float* __restrict__ W2,
                 const float* __restrict__ rad3, const float* __restrict__ cut,
                 float* __restrict__ edgeNew, float inv_sqrt_k) {
    int lane = threadIdx.x;
    int hi = lane >> 4, lr = lane & 15;
    long r = (long)blockIdx.x * 16 + lr;
    const float* Wl[3] = {W0, W1, W2};
    v8f acc[3][4];
#pragma unroll
    for (int l = 0; l < 3; ++l)
#pragma unroll
        for (int tc = 0; tc < 4; ++tc) acc[l][tc] = v8f{0,0,0,0,0,0,0,0};
    // phase 1: per-l prev-edge segment (w0 = 0 at lev0, CC otherwise)
    if (w0 > 0) {
#pragma unroll
        for (int l = 0; l < 3; ++l) {
            const float* Ar = prevE + ((long)l * EE * 2 + r) * CC;
            for (int k0 = 0; k0 < CC; k0 += 4) {
                v2f a; a.x = Ar[k0 + 2 * hi]; a.y = Ar[k0 + 2 * hi + 1];
                const float* Wp = Wl[l] + (long)(k0 + 2 * hi) * CC + lr;
#pragma unroll
                for (int tc = 0; tc < 4; ++tc) {
                    v2f b; b.x = Wp[tc * 16]; b.y = Wp[CC + tc * 16];
                    acc[l][tc] = __builtin_amdgcn_wmma_f32_16x16x4_f32(
                        false, a, false, b, (short)0, acc[l][tc], false, false);
                }
            }
        }
    }
    // phase 2: shared dots segment
    const float* Ar = dots + r * (3 * CC);
    for (int k0 = 0; k0 < w1; k0 += 4) {
        v2f a; a.x = Ar[k0 + 2 * hi]; a.y = Ar[k0 + 2 * hi + 1];
#pragma unroll
        for (int l = 0; l < 3; ++l) {
            const float* Wp = Wl[l] + (long)(w0 + k0 + 2 * hi) * CC + lr;
#pragma unroll
            for (int tc = 0; tc < 4; ++tc) {
                v2f b; b.x = Wp[tc * 16]; b.y = Wp[CC + tc * 16];
                acc[l][tc] = __builtin_amdgcn_wmma_f32_16x16x4_f32(
                    false, a, false, b, (short)0, acc[l][tc], false, false);
            }
        }
    }
    // epilogue
    long rbase = (long)blockIdx.x * 16;
#pragma unroll
    for (int v = 0; v < 8; ++v) {
        long row = rbase + v + 8 * hi;
        long e = row >> 1;
        float cu = cut[e] * inv_sqrt_k;
#pragma unroll
        for (int l = 0; l < 3; ++l) {
            const float* rl = rad3 + (long)l * EE * CC + e * CC;
            float* eo = edgeNew + (long)l * EE * 2 * CC + row * CC;
#pragma unroll
            for (int tc = 0; tc < 4; ++tc) {
                int cc2 = tc * 16 + lr;
                eo[cc2] = acc[l][tc][v] * rl[cc2] * cu;
            }
        }
    }
}

// ============================================================
// pred accumulation: pred[e] += sum_{l,c,t} edge[l][e][t][c] * Wtop[...]
// Wtop flat index = ((lev*3 + l)*C + c)*2 + t
// ============================================================
__global__ void predacc_kernel(const float* __restrict__ edge, const float* __restrict__ Wtop,
                               float* __restrict__ pred, int lev) {
    int e = blockIdx.x * blockDim.x + threadIdx.x;
    if (e >= EE) return;
    float acc = 0.0f;
    for (int l = 0; l < 3; ++l) {
        const float* E_ = edge + (long)l * EE * 2 * CC;
        const float* Wt = Wtop + ((long)(lev * 3 + l) * CC) * 2;
        for (int c = 0; c < CC; ++c) {
            acc += E_[((long)e * 2 + 0) * CC + c] * Wt[c * 2 + 0];
            acc += E_[((long)e * 2 + 1) * CC + c] * Wt[c * 2 + 1];
        }
    }
    pred[e] += acc;
}

// ============================================================
// Messages, LDS-staged sph: one block per (b,i), 64 threads (= c).
// msg[bn][s][c] = (1/N) sum_j cmul(edge_l(s)[e], sph[e][s])
// Single pass over edges; sph row cached in LDS per j.
// ============================================================
__global__ void __launch_bounds__(64)
msg_kernel(const float* __restrict__ edge, const float* __restrict__ sph,
           float* __restrict__ msg) {
    __shared__ float S[18];
    int bn = blockIdx.x;
    int c = threadIdx.x;
    float re[9], im[9];
#pragma unroll
    for (int s = 0; s < 9; ++s) { re[s] = 0.0f; im[s] = 0.0f; }
    for (int j = 0; j < NN; ++j) {
        long e = (long)bn * NN + j;
        __syncthreads();
        if (c < 18) S[c] = sph[e * 18 + c];
        __syncthreads();
        float er[3], ei[3];
#pragma unroll
        for (int l = 0; l < 3; ++l) {
            const float* E_ = edge + (long)l * EE * 2 * CC;
            er[l] = E_[(e * 2 + 0) * CC + c];
            ei[l] = E_[(e * 2 + 1) * CC + c];
        }
#pragma unroll
        for (int s = 0; s < 9; ++s) {
            int l = (s >= 4) ? 2 : ((s >= 1) ? 1 : 0);
            float sr = S[s * 2], si = S[s * 2 + 1];
            re[s] += er[l] * sr - ei[l] * si;
            im[s] += er[l] * si + ei[l] * sr;
        }
    }
#pragma unroll
    for (int s = 0; s < 9; ++s) {
        long o = (((long)bn * 9 + s) * CC + c) * 2;
        msg[o]     = re[s] / (float)NN;
        msg[o + 1] = im[s] / (float)NN;
    }
}

// ============================================================
// CG feature builder. cgf layout: [L][bn][m(5)][t(2)][512], d = p*C + c.
// Part order per reference: atom[L] (if present), msg[L], then CG(l1,l2)
// for l1 in atom, l2 in 0..2 with |l1-l2| <= L <= l1+l2.
// ============================================================
__global__ void cgfeat_kernel(const float* __restrict__ atom, const float* __restrict__ msg,
                              const float* __restrict__ cgtab, float* __restrict__ cgf, int n_l) {
    int gid = blockIdx.x * blockDim.x + threadIdx.x;
    if (gid >= 3 * BN * CC) return;
    int c = gid % CC;
    int bn = (gid / CC) % BN;
    int L = gid / (CC * BN);
    float* out = cgf + ((long)L * BN + bn) * 5 * 2 * 512;
    const float* A  = atom + (long)bn * 9 * CC * 2;
    const float* Mg = msg + (long)bn * 9 * CC * 2;
    int p = 0;
    if (L < n_l) {
        for (int m = 0; m <= 2 * L; ++m) {
            long ai = ((long)(L * L + m) * CC + c) * 2;
            out[((long)m * 2 + 0) * 512 + p * CC + c] = A[ai];
            out[((long)m * 2 + 1) * 512 + p * CC + c] = A[ai + 1];
        }
        p++;
    }
    for (int m = 0; m <= 2 * L; ++m) {
        long mi = ((long)(L * L + m) * CC + c) * 2;
        out[((long)m * 2 + 0) * 512 + p * CC + c] = Mg[mi];
        out[((long)m * 2 + 1) * 512 + p * CC + c] = Mg[mi + 1];
    }
    p++;
    for (int l1 = 0; l1 < n_l; ++l1) {
        for (int l2 = 0; l2 <= 2; ++l2) {
            int dl = (l1 > l2) ? (l1 - l2) : (l2 - l1);
            if (dl > L || L > l1 + l2) continue;
            for (int mo = 0; mo <= 2 * L; ++mo) {
                float re = 0.0f, im = 0.0f;
                for (int m1 = 0; m1 <= 2 * l1; ++m1) {
                    for (int m2 = 0; m2 <= 2 * l2; ++m2) {
                        float w = cgtab[((((l1 * 3 + l2) * 3 + L) * 5 + m1) * 5 + m2) * 5 + mo];
                        if (w == 0.0f) continue;
                        long ai = ((long)(l1 * l1 + m1) * CC + c) * 2;
                        long bi = ((long)(l2 * l2 + m2) * CC + c) * 2;
                        float ar = A[ai], aii = A[ai + 1];
                        float br = Mg[bi], bii = Mg[bi + 1];
                        re += w * (ar * br - aii * bii);
                        im += w * (ar * bii + aii * br);
                    }
                }
                out[((long)mo * 2 + 0) * 512 + p * CC + c] = re;
                out[((long)mo * 2 + 1) * 512 + p * CC + c] = im;
            }
            p++;
        }
    }
}

// ============================================================
// Atom mixing: atom_new[bn][L*L+m][c][t] =
//   (1/sqrt(k)) * sum_d cgf[L][bn][m][t][d] * Watom[lev][L][d][c] * mask
// ============================================================
__global__ void atommix_kernel(const float* __restrict__ cgf, const float* __restrict__ Watom,
                               const float* __restrict__ am, float* __restrict__ atom_new,
                               int n_l, int lev) {
    int gid = blockIdx.x * blockDim.x + threadIdx.x;
    if (gid >= 3 * BN * CC * 10) return;
    int t = gid % 2;
    int m = (gid / 2) % 5;
    int c = (gid / 10) % CC;
    int bn = (gid / (10 * CC)) % BN;
    int L = gid / (10 * CC * BN);
    if (m > 2 * L) return;
    int np = (L < n_l) ? 2 : 1;
    for (int l1 = 0; l1 < n_l; ++l1)
        for (int l2 = 0; l2 <= 2; ++l2) {
            int dl = (l1 > l2) ? (l1 - l2) : (l2 - l1);
            if (dl <= L && L <= l1 + l2) np++;
        }
    int k = np * CC;
    const float* f = cgf + (((long)L * BN + bn) * 5 + m) * 2 * 512 + (long)t * 512;
    const float* W = Watom + ((long)(lev * 3 + L) * 512) * CC + c;
    float acc = 0.0f;
    for (int d = 0; d < k; ++d) acc += f[d] * W[(long)d * CC];
    atom_new[(((long)bn * 9 + L * L + m) * CC + c) * 2 + t] =
        acc * (1.0f / sqrtf((float)k)) * am[bn];
}

// ============================================================
// Final: out[e] = (pred[e] + b_top) * edge_mask
// ============================================================
__global__ void final_kernel(const float* __restrict__ pred, const float* __restrict__ am,
                             const float* __restrict__ btop, float* __restrict__ out) {
    int e = blockIdx.x * blockDim.x + threadIdx.x;
    if (e >= EE) return;
    int j = e % NN, bi = e / NN;
    int i = bi % NN, b = bi / NN;
    out[e] = (pred[e] + btop[0]) * am[b * NN + i] * am[b * NN + j];
}

// ============================================================
// Host orchestration (graph-capturable: all launches on stream,
// no alloc/sync, deterministic).
// ============================================================
extern "C" void kernel_launch(void* const* d_in, const int* in_sizes, int n_in,
                              void* d_out, int out_size, void* d_ws, size_t ws_size,
                              hipStream_t stream) {
    (void)in_sizes; (void)n_in; (void)out_size; (void)ws_size;
    const float* pos     = (const float*)d_in[0];
    const float* am      = (const float*)d_in[1];
    const int*   species = (const int*)d_in[2];
    const int*   charges = (const int*)d_in[3];
    const float* Win     = (const float*)d_in[4];   // [15][64]
    const float* Wrad    = (const float*)d_in[5];   // [3][3][32][64]
    const float* Wedge0  = (const float*)d_in[6];   // [3][64][64]
    const float* Wedger  = (const float*)d_in[7];   // [2][3][256][64]
    const float* Watom   = (const float*)d_in[8];   // [3][3][512][64]
    const float* Wtop    = (const float*)d_in[9];   // [1152][1]
    const float* btop    = (const float*)d_in[10];  // [1]

    float* ws = (float*)d_ws;
    size_t off = 0;
    float* sph   = ws + off; off += (size_t)EE * 18;
    float* cut   = ws + off; off += (size_t)EE;
    float* basis = ws + off; off += (size_t)EE * NBAS;
    float* rad3  = ws + off; off += (size_t)3 * EE * CC;
    float* atomA = ws + off; off += (size_t)BN * 9 * CC * 2;
    float* atomB = ws + off; off += (size_t)BN * 9 * CC * 2;
    float* msg   = ws + off; off += (size_t)BN * 9 * CC * 2;
    float* dots  = ws + off; off += (size_t)EE * 2 * 3 * CC;
    float* edgeA = ws + off; off += (size_t)3 * EE * 2 * CC;
    float* edgeB = ws + off; off += (size_t)3 * EE * 2 * CC;
    float* cgf   = ws + off; off += (size_t)3 * BN * 5 * 2 * 512;
    float* cgtab = ws + off; off += 3 * 3 * 3 * 5 * 5 * 5;
    float* pred  = ws + off; off += (size_t)EE;

    cg_init_kernel<<<(3375 + 255) / 256, 256, 0, stream>>>(cgtab);
    geom_kernel<<<(EE + 255) / 256, 256, 0, stream>>>(pos, am, sph, cut, basis, pred);
    input_kernel<<<(BN * CC + 255) / 256, 256, 0, stream>>>(species, charges, am, Win, atomA);

    float* atomCur = atomA; float* atomNew = atomB;
    float* edgePrev = edgeA; float* edgeNew = edgeB;
    int n_l = 1;
    for (int lev = 0; lev < 3; ++lev) {
        dots_kernel<<<(EE * CC + 255) / 256, 256, 0, stream>>>(atomCur, dots, n_l);
        // radial GEMMs for all 3 l of this level
        {
            const float* WradLev = Wrad + (size_t)lev * 3 * NBAS * CC;
            dim3 g(EE / 16, 3);
            rad3_wmma_kernel<<<g, 32, 0, stream>>>(basis, WradLev, rad3);
        }
        // fused edge mixing for all 3 l
        if (lev == 0) {
            mix3_wmma_kernel<<<(2 * EE) / 16, 32, 0, stream>>>(
                dots /*unused*/, 0, dots, CC,
                Wedge0 + 0 * CC * CC, Wedge0 + 1 * CC * CC, Wedge0 + 2 * CC * CC,
                rad3, cut, edgeNew, 0.125f /* 1/sqrt(64) */);
        } else {
            const float* Wb = Wedger + (size_t)(lev - 1) * 3 * 4 * CC * CC;
            mix3_wmma_kernel<<<(2 * EE) / 16, 32, 0, stream>>>(
                edgePrev, CC, dots, 3 * CC,
                Wb + 0 * (size_t)4 * CC * CC, Wb + 1 * (size_t)4 * CC * CC,
                Wb + 2 * (size_t)4 * CC * CC,
                rad3, cut, edgeNew, 0.0625f /* 1/sqrt(256) */);
        }
        predacc_kernel<<<(EE + 255) / 256, 256, 0, stream>>>(edgeNew, Wtop, pred, lev);
        msg_kernel<<<BN, 64, 0, stream>>>(edgeNew, sph, msg);
        cgfeat_kernel<<<(3 * BN * CC + 255) / 256, 256, 0, stream>>>(atomCur, msg, cgtab, cgf, n_l);
        atommix_kernel<<<(3 * BN * CC * 10 + 255) / 256, 256, 0, stream>>>(
            cgf, Watom, am, atomNew, n_l, lev);
        { float* t = atomCur; atomCur = atomNew; atomNew = t; }
        { float* t = edgePrev; edgePrev = edgeNew; edgeNew = t; }
        n_l = 3;
    }
    final_kernel<<<(EE + 255) / 256, 256, 0, stream>>>(pred, am, btop, (float*)d_out);
}